// GraphAttentionLayer_36867999269042
// MI455X (gfx1250) — compile-verified
//
#include <hip/hip_runtime.h>

typedef __attribute__((ext_vector_type(2))) float v2f;
typedef __attribute__((ext_vector_type(4))) float v4f;
typedef __attribute__((ext_vector_type(8))) float v8f;

#define D 128
#define GAT_ALPHA 0.2f

// ---------------------------------------------------------------------------
// Kernel 1: Wh = h @ W^T using V_WMMA_F32_16X16X4_F32 (fp32 matrix pipe).
// Block = 256 threads (8 waves). Each wave computes a 16-row x 128-col strip.
// W (128x128 f32 = 64KB) is staged in LDS with a +2*row element skew so the
// B-fragment ds_load_b64 (16 lanes, 512B row stride) is bank-conflict-free.
// N is a multiple of 16, so all WMMA tiles are full (EXEC all-ones).
// ---------------------------------------------------------------------------
__global__ __launch_bounds__(256) void gat_gemm_wmma(
    const float* __restrict__ h, const float* __restrict__ W,
    float* __restrict__ Wh, int N)
{
    __shared__ float Wlds[D * D];  // element (n,k) stored at n*128 + ((k+2n)&127)

    const int tid = threadIdx.x;

    // Cooperative skewed fill: 16384 floats / 256 threads = 16 float4 each.
    for (int i = tid; i < (D * D) / 4; i += 256) {
        const int row = i >> 5;           // 32 float4 per row
        const int k   = (i & 31) << 2;
        const v4f v   = *(const v4f*)(W + row * D + k);
        const int base = row * D;
        const int sk   = 2 * row;
        Wlds[base + ((k + 0 + sk) & 127)] = v.x;
        Wlds[base + ((k + 1 + sk) & 127)] = v.y;
        Wlds[base + ((k + 2 + sk) & 127)] = v.z;
        Wlds[base + ((k + 3 + sk) & 127)] = v.w;
    }
    __syncthreads();

    const int wave = tid >> 5;
    const int lane = tid & 31;
    const int m0   = blockIdx.x * 128 + wave * 16;
    if (m0 >= N) return;                  // uniform per wave; active waves keep EXEC full

    const int half = lane >> 4;           // 0: K,K+1 ; 1: K+2,K+3 (fp32 A layout)
    const int lr   = lane & 15;
    const int arow = m0 + lr;

    v8f acc[8] = {};                      // 8 N-tiles of 16 columns, all zero-init

    const float* hrow = h + (size_t)arow * D + 2 * half;

    for (int k0 = 0; k0 < D; k0 += 4) {
        const v2f afrag = *(const v2f*)(hrow + k0);
        const int kk = k0 + 2 * half;
#pragma unroll
        for (int t = 0; t < 8; ++t) {
            const int n = t * 16 + lr;    // B column index held by this lane
            const v2f bfrag = *(const v2f*)&Wlds[n * D + ((kk + 2 * n) & 127)];
            acc[t] = __builtin_amdgcn_wmma_f32_16x16x4_f32(
                false, afrag, false, bfrag, (short)0, acc[t], false, false);
        }
    }

    // C/D layout: VGPR j holds M = m0 + j + 8*half, N = t*16 + lr.
#pragma unroll
    for (int t = 0; t < 8; ++t) {
#pragma unroll
        for (int j = 0; j < 8; ++j) {
            const int m = m0 + j + 8 * half;
            Wh[(size_t)m * D + t * 16 + lr] = acc[t][j];
        }
    }
}

// ---------------------------------------------------------------------------
// Kernel 0: zero the output accumulator, seed softmax max/sum scalars.
// ---------------------------------------------------------------------------
__global__ __launch_bounds__(256) void gat_init(
    float* __restrict__ out, int n, float* __restrict__ gmax, float* __restrict__ gsum)
{
    const int i = blockIdx.x * 256 + threadIdx.x;
    if (i < n) out[i] = 0.0f;
    if (i == 0) { *gmax = -__builtin_inff(); *gsum = 0.0f; }
}

// ---------------------------------------------------------------------------
// Kernel 2: s_l[i] = Wh[i,:].a_l ; s_r[i] = Wh[i,:].a_r  (one wave per row).
// ---------------------------------------------------------------------------
__global__ __launch_bounds__(256) void gat_rowdot(
    const float* __restrict__ Wh, const float* __restrict__ a,
    float* __restrict__ s_l, float* __restrict__ s_r, int N)
{
    const int row  = (blockIdx.x * 256 + threadIdx.x) >> 5;
    const int lane = threadIdx.x & 31;
    if (row >= N) return;

    const v4f w  = *(const v4f*)(Wh + (size_t)row * D + lane * 4);
    const v4f al = *(const v4f*)(a + lane * 4);
    const v4f ar = *(const v4f*)(a + D + lane * 4);

    float pl = w.x * al.x + w.y * al.y + w.z * al.z + w.w * al.w;
    float pr = w.x * ar.x + w.y * ar.y + w.z * ar.z + w.w * ar.w;

#pragma unroll
    for (int off = 16; off > 0; off >>= 1) {
        pl += __shfl_xor(pl, off, 32);
        pr += __shfl_xor(pr, off, 32);
    }
    if (lane == 0) { s_l[row] = pl; s_r[row] = pr; }
}

__device__ __forceinline__ float atomicMaxF(float* addr, float val)
{
    int* ia  = (int*)addr;
    int  old = __hip_atomic_load(ia, __ATOMIC_RELAXED, __HIP_MEMORY_SCOPE_AGENT);
    while (__int_as_float(old) < val) {
        const int assumed = old;
        old = atomicCAS(ia, assumed, __float_as_int(val));
        if (old == assumed) break;
    }
    return __int_as_float(old);
}

// ---------------------------------------------------------------------------
// Kernel 3: e[k] = leakyrelu(s_l[row[k]] + s_r[col[k]]), global max via atomics.
// ---------------------------------------------------------------------------
__global__ __launch_bounds__(256) void gat_edge_logits(
    const int* __restrict__ ei, const float* __restrict__ s_l,
    const float* __restrict__ s_r, float* __restrict__ p,
    float* __restrict__ gmax, int E)
{
    __shared__ float red[8];
    const int k = blockIdx.x * 256 + threadIdx.x;

    float e = -__builtin_inff();
    if (k < E) {
        const int r = ei[k];
        const int c = ei[E + k];
        const float v = s_l[r] + s_r[c];
        e = (v >= 0.0f) ? v : GAT_ALPHA * v;
        p[k] = e;
    }
#pragma unroll
    for (int off = 16; off > 0; off >>= 1) e = fmaxf(e, __shfl_xor(e, off, 32));
    if ((threadIdx.x & 31) == 0) red[threadIdx.x >> 5] = e;
    __syncthreads();
    if (threadIdx.x < 8) {
        float v = red[threadIdx.x];
#pragma unroll
        for (int off = 4; off > 0; off >>= 1) v = fmaxf(v, __shfl_xor(v, off, 8));
        if (threadIdx.x == 0) atomicMaxF(gmax, v);
    }
}

// ---------------------------------------------------------------------------
// Kernel 4: p[k] = exp(e[k] - max), accumulate global denominator.
// ---------------------------------------------------------------------------
__global__ __launch_bounds__(256) void gat_edge_exp(
    float* __restrict__ p, const float* __restrict__ gmax,
    float* __restrict__ gsum, int E)
{
    __shared__ float red[8];
    const int k = blockIdx.x * 256 + threadIdx.x;
    const float m = *gmax;

    float v = 0.0f;
    if (k < E) {
        v = __expf(p[k] - m);
        p[k] = v;
    }
#pragma unroll
    for (int off = 16; off > 0; off >>= 1) v += __shfl_xor(v, off, 32);
    if ((threadIdx.x & 31) == 0) red[threadIdx.x >> 5] = v;
    __syncthreads();
    if (threadIdx.x < 8) {
        float s = red[threadIdx.x];
#pragma unroll
        for (int off = 4; off > 0; off >>= 1) s += __shfl_xor(s, off, 8);
        if (threadIdx.x == 0) atomicAdd(gsum, s);
    }
}

// ---------------------------------------------------------------------------
// Kernel 5: out[row[k],:] += (p[k]/sum) * Wh[col[k],:]   (one wave per edge,
// float4 per lane, native GLOBAL_ATOMIC_ADD_F32 scatter).
// ---------------------------------------------------------------------------
__global__ __launch_bounds__(256) void gat_scatter(
    const int* __restrict__ ei, const float* __restrict__ p,
    const float* __restrict__ gsum, const float* __restrict__ Wh,
    float* __restrict__ out, int E)
{
    const int edge = (blockIdx.x * 256 + threadIdx.x) >> 5;
    const int lane = threadIdx.x & 31;
    if (edge >= E) return;

    const float coeff = p[edge] * __frcp_rn(*gsum);
    const int r = ei[edge];
    const int c = ei[E + edge];

    const v4f w = *(const v4f*)(Wh + (size_t)c * D + lane * 4);
    float* dst = out + (size_t)r * D + lane * 4;
    atomicAdd(dst + 0, coeff * w.x);
    atomicAdd(dst + 1, coeff * w.y);
    atomicAdd(dst + 2, coeff * w.z);
    atomicAdd(dst + 3, coeff * w.w);
}

// ---------------------------------------------------------------------------
// Launch: h[0], edge_index[1], W[2], a[3] -> d_out = h_prime (N x 128 f32).
// Workspace: Wh (N*128) | s_l (N) | s_r (N) | p (E) | gmax | gsum
// ---------------------------------------------------------------------------
extern "C" void kernel_launch(void* const* d_in, const int* in_sizes, int n_in,
                              void* d_out, int out_size, void* d_ws, size_t ws_size,
                              hipStream_t stream)
{
    const float* h  = (const float*)d_in[0];
    const int*   ei = (const int*)d_in[1];
    const float* W  = (const float*)d_in[2];
    const float* a  = (const float*)d_in[3];

    const int N = in_sizes[0] / D;
    const int E = in_sizes[1] / 2;

    float* out  = (float*)d_out;
    float* ws   = (float*)d_ws;
    float* Wh   = ws;
    float* s_l  = Wh + (size_t)N * D;
    float* s_r  = s_l + N;
    float* p    = s_r + N;
    float* gmax = p + E;
    float* gsum = gmax + 1;

    gat_init<<<(out_size + 255) / 256, 256, 0, stream>>>(out, out_size, gmax, gsum);
    gat_gemm_wmma<<<(N + 127) / 128, 256, 0, stream>>>(h, W, Wh, N);
    gat_rowdot<<<(N + 7) / 8, 256, 0, stream>>>(Wh, a, s_l, s_r, N);
    gat_edge_logits<<<(E + 255) / 256, 256, 0, stream>>>(ei, s_l, s_r, p, gmax, E);
    gat_edge_exp<<<(E + 255) / 256, 256, 0, stream>>>(p, gmax, gsum, E);
    gat_scatter<<<(E + 7) / 8, 256, 0, stream>>>(ei, p, gsum, Wh, out, E);
}